// LSNet_7584912245499
// MI455X (gfx1250) — compile-verified
//
#include <hip/hip_runtime.h>
#include <hip/hip_bf16.h>
#include <math.h>

// ---------------------------------------------------------------------------
// LSNet weighted least-squares fit:
//   w = softmax(weights)[lbl];  G = [X|F]^T diag(w^2) [X|F] pieces
//   out = (inv(XtX) @ XtF)^T per batch.
// Memory-bound: ~154 MB @ 23.3 TB/s => ~6.6us floor. Main loop is f32 VALU
// streaming (one point/lane); WMMA f32 16x16x4 (ones-matrix column-sum) is
// used for the free cross-wave block reduction.
// ---------------------------------------------------------------------------

typedef __attribute__((ext_vector_type(2))) float v2f;
typedef __attribute__((ext_vector_type(8))) float v8f;

#define BS      64
#define NP      100000
#define NC      20
#define CHUNKS  16
#define CH      (NP / CHUNKS)     // 6250
#define TPB     256
#define NACC    12
#define WS_SUM_OFF 32             // float offset of accumulator region in ws

__global__ void lsnet_init_ws(float* __restrict__ ws) {
    int gid = blockIdx.x * blockDim.x + threadIdx.x;
    if (gid < BS * NACC) ws[WS_SUM_OFF + gid] = 0.0f;
}

__global__ void lsnet_reduce(const float* __restrict__ X,
                             const float* __restrict__ Fm,
                             const float* __restrict__ W,
                             const int*   __restrict__ lbl,
                             float*       __restrict__ ws) {
    __shared__ float wraw[NC];
    __shared__ float wexpv[NC];
    __shared__ float prob[NC];
    __shared__ float wsum[8][16];

    const int tid = threadIdx.x;

    // --- per-block softmax table (20 entries; redundant + cheap) ---
    if (tid < NC) wraw[tid] = W[tid];
    __syncthreads();
    if (tid < NC) {
        float m = -1e30f;
        #pragma unroll
        for (int c = 0; c < NC; ++c) m = fmaxf(m, wraw[c]);
        wexpv[tid] = expf(wraw[tid] - m);
    }
    __syncthreads();
    if (tid < NC) {
        float s = 0.0f;
        #pragma unroll
        for (int c = 0; c < NC; ++c) s += wexpv[c];
        prob[tid] = wexpv[tid] / s;
    }
    __syncthreads();

    // --- streaming accumulation: one point per lane per iteration ---
    const int b = blockIdx.x / CHUNKS;
    const int c = blockIdx.x % CHUNKS;
    const size_t base = (size_t)b * NP + (size_t)c * CH;

    float acc[NACC];
    #pragma unroll
    for (int j = 0; j < NACC; ++j) acc[j] = 0.0f;

    for (int i = tid; i < CH; i += TPB) {
        const size_t p = base + (size_t)i;
        const float* xp = X + p * 3;
        const float x = xp[0], y = xp[1], z = xp[2];
        const float* fp = Fm + p * 2;
        const float f0 = fp[0], f1 = fp[1];
        const int   l  = lbl[p];
        const float w  = prob[l];
        const float ww = w * w;
        const float ax = ww * x, ay = ww * y, az = ww * z;
        acc[0]  += ax * x;  acc[1]  += ax * y;  acc[2]  += ax * z;
        acc[3]  += ay * y;  acc[4]  += ay * z;  acc[5]  += az * z;
        acc[6]  += ax * f0; acc[7]  += ax * f1;
        acc[8]  += ay * f0; acc[9]  += ay * f1;
        acc[10] += az * f0; acc[11] += az * f1;
    }

    // --- intra-wave reduction (wave32) ---
    #pragma unroll
    for (int j = 0; j < NACC; ++j) {
        float v = acc[j];
        #pragma unroll
        for (int off = 16; off > 0; off >>= 1)
            v += __shfl_down(v, off, 32);
        acc[j] = v;
    }

    const int wave = tid >> 5;
    const int lane = tid & 31;
    if (lane == 0) {
        #pragma unroll
        for (int j = 0; j < NACC; ++j) wsum[wave][j] = acc[j];
        #pragma unroll
        for (int j = NACC; j < 16; ++j) wsum[wave][j] = 0.0f;
    }
    __syncthreads();

    // --- cross-wave reduction via WMMA: D = ones(16x4) x B(4x16) => colsums.
    // A == all-ones makes the result invariant to the K-row <-> VGPR-half
    // assignment of the 32-bit B layout; only N = lane&15 matters.
    if (tid < 32) {
        const int n = tid & 15;
        const int g = (tid < 16) ? 0 : 2;

        v2f A; A.x = 1.0f; A.y = 1.0f;

        v2f B0; B0.x = wsum[g + 0][n]; B0.y = wsum[g + 1][n];
        v8f Cd = {};
        Cd = __builtin_amdgcn_wmma_f32_16x16x4_f32(
                false, A, false, B0, (short)0, Cd, false, false);

        v2f B1; B1.x = wsum[g + 4][n]; B1.y = wsum[g + 5][n];
        Cd = __builtin_amdgcn_wmma_f32_16x16x4_f32(
                false, A, false, B1, (short)0, Cd, false, false);

        // D vgpr0, lanes 0..15 hold (M=0, N=lane) => lane n has colsum[n].
        if (tid < NACC) {
            atomicAdd(&ws[WS_SUM_OFF + b * NACC + tid], Cd[0]);
        }
    }
}

__global__ void lsnet_solve(const float* __restrict__ ws,
                            float*       __restrict__ out) {
    const int b = blockIdx.x * blockDim.x + threadIdx.x;
    if (b >= BS) return;
    const float* s = ws + WS_SUM_OFF + b * NACC;
    const float xx = s[0], xy = s[1], xz = s[2];
    const float yy = s[3], yz = s[4], zz = s[5];

    // adjugate of symmetric 3x3
    const float a00 = yy * zz - yz * yz;
    const float a01 = xz * yz - xy * zz;
    const float a02 = xy * yz - xz * yy;
    const float a11 = xx * zz - xz * xz;
    const float a12 = xy * xz - xx * yz;
    const float a22 = xx * yy - xy * xy;
    const float det = xx * a00 + xy * a01 + xz * a02;
    const float inv = 1.0f / det;
    const float i00 = a00 * inv, i01 = a01 * inv, i02 = a02 * inv;
    const float i11 = a11 * inv, i12 = a12 * inv, i22 = a22 * inv;

    const float xf0 = s[6],  xf1 = s[7];
    const float yf0 = s[8],  yf1 = s[9];
    const float zf0 = s[10], zf1 = s[11];

    // T = inv(XtX) @ XtF  (3x2)
    const float t00 = i00 * xf0 + i01 * yf0 + i02 * zf0;
    const float t01 = i00 * xf1 + i01 * yf1 + i02 * zf1;
    const float t10 = i01 * xf0 + i11 * yf0 + i12 * zf0;
    const float t11 = i01 * xf1 + i11 * yf1 + i12 * zf1;
    const float t20 = i02 * xf0 + i12 * yf0 + i22 * zf0;
    const float t21 = i02 * xf1 + i12 * yf1 + i22 * zf1;

    // output is swapaxes -> [bs, 2, 3]: out[b][j][i] = T[i][j]
    float* o = out + b * 6;
    o[0] = t00; o[1] = t10; o[2] = t20;
    o[3] = t01; o[4] = t11; o[5] = t21;
}

extern "C" void kernel_launch(void* const* d_in, const int* in_sizes, int n_in,
                              void* d_out, int out_size, void* d_ws, size_t ws_size,
                              hipStream_t stream) {
    const float* X   = (const float*)d_in[0];
    const float* Fm  = (const float*)d_in[1];
    const float* W   = (const float*)d_in[2];
    const int*   lbl = (const int*)d_in[3];
    float*       ws  = (float*)d_ws;
    float*       out = (float*)d_out;
    (void)in_sizes; (void)n_in; (void)out_size; (void)ws_size;

    lsnet_init_ws<<<3, TPB, 0, stream>>>(ws);
    lsnet_reduce<<<BS * CHUNKS, TPB, 0, stream>>>(X, Fm, W, lbl, ws);
    lsnet_solve<<<1, BS, 0, stream>>>(ws, out);
}